// HealEncoding_77154792505522
// MI455X (gfx1250) — compile-verified
//
#include <hip/hip_runtime.h>
#include <stdint.h>

// ---------------- problem constants (from reference) ----------------
#define NLVL   8
#define FDIM   8
#define NPIX   196608        // 12 * 128 * 128
#define NBATCH 1000000

// ---------------- kernel tuning ----------------
#define TPB          128     // 4 wave32 waves per block
#define LDS_LVLS     3       // levels 0..2 cached in LDS (12+48+192 = 252 rows = 8064 B)
#define TBL_DW       2016    // 252 rows * 8 floats
#define STAGE_STRIDE 68      // padded dwords per batch element (64 used + 4 pad) -> low bank conflicts

typedef float v4f __attribute__((ext_vector_type(4)));
typedef int   v4i __attribute__((ext_vector_type(4)));

#if __has_builtin(__builtin_amdgcn_global_load_async_to_lds_b128) && \
    __has_builtin(__builtin_amdgcn_s_wait_asynccnt)
#define HAVE_ASYNC_LDS 1
// builtin signature (from hipcc diagnostic): (v4i AS1*, v4i AS3*, imm offset, imm cpol)
typedef __attribute__((address_space(1))) v4i* gbl_v4i_p;
typedef __attribute__((address_space(3))) v4i* lds_v4i_p;
#else
#define HAVE_ASYNC_LDS 0
#endif

__device__ __forceinline__ constexpr int lds_base_dw(int l) {
  // 12*(4^l - 1)/3 rows * 8 floats
  return (l == 0) ? 0 : (l == 1) ? 96 : 480;
}

__global__ __launch_bounds__(TPB) void heal_encoding_kernel(
    const float* __restrict__ params,        // [8][196608][8] f32
    const int*   __restrict__ neigh_pix,     // [8][4][1e6]    i32
    const float* __restrict__ neigh_weight,  // [8][4][1e6]    f32
    float*       __restrict__ out)           // [1e6][64]      f32, out[b, f*8+l]
{
  __shared__ float s_table[TBL_DW];                 //  8,064 B : levels 0..2 tables
  __shared__ float s_stage[TPB * STAGE_STRIDE];     // 34,816 B : output transpose staging

  const int t = threadIdx.x;

  // ---------------- Phase A: cache small level tables in LDS ----------------
  {
    const v4f* g0 = reinterpret_cast<const v4f*>(params);                        //  24 vec4
    const v4f* g1 = reinterpret_cast<const v4f*>(params + (size_t)NPIX * 8);     //  96 vec4
    const v4f* g2 = reinterpret_cast<const v4f*>(params + (size_t)2 * NPIX * 8); // 384 vec4
    v4f* l0 = reinterpret_cast<v4f*>(&s_table[0]);
    v4f* l1 = reinterpret_cast<v4f*>(&s_table[96]);
    v4f* l2 = reinterpret_cast<v4f*>(&s_table[480]);
#if HAVE_ASYNC_LDS
    if (t < 24)
      __builtin_amdgcn_global_load_async_to_lds_b128(
          (gbl_v4i_p)(g0 + t), (lds_v4i_p)(l0 + t), 0, 0);
    if (t < 96)
      __builtin_amdgcn_global_load_async_to_lds_b128(
          (gbl_v4i_p)(g1 + t), (lds_v4i_p)(l1 + t), 0, 0);
#pragma unroll
    for (int k = 0; k < 3; ++k)
      __builtin_amdgcn_global_load_async_to_lds_b128(
          (gbl_v4i_p)(g2 + k * TPB + t), (lds_v4i_p)(l2 + k * TPB + t), 0, 0);
    __builtin_amdgcn_s_wait_asynccnt(0);
#else
    if (t < 24) l0[t] = g0[t];
    if (t < 96) l1[t] = g1[t];
#pragma unroll
    for (int k = 0; k < 3; ++k) l2[k * TPB + t] = g2[k * TPB + t];
#endif
  }
  __syncthreads();

  // ---------------- Phase B: per-point gather + weighted accumulate ----------------
  const int b = blockIdx.x * TPB + t;

  float acc[64];  // acc[f*8 + l] -- matches output interleave
#pragma unroll
  for (int i = 0; i < 64; ++i) acc[i] = 0.0f;

  if (b < NBATCH) {
#pragma unroll
    for (int l = 0; l < NLVL; ++l) {
      const int*   ip = neigh_pix    + ((size_t)l * 4) * NBATCH + b;
      const float* wp = neigh_weight + ((size_t)l * 4) * NBATCH + b;
#pragma unroll
      for (int n = 0; n < 4; ++n) {
        // stream-once data: non-temporal (keep L2 for the gather tables)
        const int   pix = __builtin_nontemporal_load(ip + (size_t)n * NBATCH);
        const float w   = __builtin_nontemporal_load(wp + (size_t)n * NBATCH);
        v4f r0, r1;
        if (l < LDS_LVLS) {  // compile-time branch (loop fully unrolled)
          const v4f* rr = reinterpret_cast<const v4f*>(&s_table[lds_base_dw(l) + pix * 8]);
          r0 = rr[0];
          r1 = rr[1];
        } else {
          const v4f* rr = reinterpret_cast<const v4f*>(
              params + ((size_t)l * NPIX + (size_t)pix) * 8);
          r0 = rr[0];
          r1 = rr[1];
        }
        acc[0 * 8 + l] = fmaf(w, r0.x, acc[0 * 8 + l]);
        acc[1 * 8 + l] = fmaf(w, r0.y, acc[1 * 8 + l]);
        acc[2 * 8 + l] = fmaf(w, r0.z, acc[2 * 8 + l]);
        acc[3 * 8 + l] = fmaf(w, r0.w, acc[3 * 8 + l]);
        acc[4 * 8 + l] = fmaf(w, r1.x, acc[4 * 8 + l]);
        acc[5 * 8 + l] = fmaf(w, r1.y, acc[5 * 8 + l]);
        acc[6 * 8 + l] = fmaf(w, r1.z, acc[6 * 8 + l]);
        acc[7 * 8 + l] = fmaf(w, r1.w, acc[7 * 8 + l]);
      }
    }
    // stage this point's 64 outputs in LDS (padded row -> 16B aligned, low conflict)
    v4f* sp = reinterpret_cast<v4f*>(&s_stage[t * STAGE_STRIDE]);
#pragma unroll
    for (int k = 0; k < 16; ++k) {
      v4f v;
      v.x = acc[4 * k + 0];
      v.y = acc[4 * k + 1];
      v.z = acc[4 * k + 2];
      v.w = acc[4 * k + 3];
      sp[k] = v;
    }
  }
  __syncthreads();

  // ---------------- Phase C: coalesced non-temporal writeout ----------------
  // Block covers TPB*64 = 8192 output dwords; 128 threads * vec4 = 512 dwords/iter.
  const int rem      = NBATCH - blockIdx.x * TPB;
  const int valid_dw = (rem >= TPB ? TPB : rem) * 64;
  float* outp = out + (size_t)blockIdx.x * (TPB * 64);
#pragma unroll
  for (int i = 0; i < 16; ++i) {
    const int q = i * (TPB * 4) + t * 4;  // local dword index
    if (q < valid_dw) {
      const int bb = q >> 6;   // batch element within block
      const int c  = q & 63;   // dword within element
      const v4f v = *reinterpret_cast<const v4f*>(&s_stage[bb * STAGE_STRIDE + c]);
      __builtin_nontemporal_store(v, reinterpret_cast<v4f*>(outp + q));
    }
  }
}

extern "C" void kernel_launch(void* const* d_in, const int* in_sizes, int n_in,
                              void* d_out, int out_size, void* d_ws, size_t ws_size,
                              hipStream_t stream) {
  const float* params       = (const float*)d_in[0];
  const int*   neigh_pix    = (const int*)d_in[1];
  const float* neigh_weight = (const float*)d_in[2];
  float*       out          = (float*)d_out;

  const int grid = (NBATCH + TPB - 1) / TPB;  // 7813 blocks
  hipLaunchKernelGGL(heal_encoding_kernel, dim3(grid), dim3(TPB), 0, stream,
                     params, neigh_pix, neigh_weight, out);
}